// HGCNConv_56788057588086
// MI455X (gfx1250) — compile-verified
//
#include <hip/hip_runtime.h>
#include <hip/hip_bf16.h>

// ---------------------------------------------------------------------------
// Hyperbolic GCN conv (Poincare ball, c=1) for MI455X / gfx1250 (wave32).
//   1) mx = x @ W^T via V_WMMA_F32_16X16X4_F32; one wave computes a full
//      16x96 output strip (A loaded once, 6 independent accumulators)
//   2) per-row hyperbolic epilogue (mobius_matvec tail + bias mobius_add + logmap0)
//   3) edge scatter (spmm) with global_atomic_add_f32 (L2-resident tables)
//   4) expmap0 -> proj -> logmap0 -> relu -> expmap0 -> proj
// ---------------------------------------------------------------------------

typedef __attribute__((ext_vector_type(2))) float v2f;
typedef __attribute__((ext_vector_type(8))) float v8f;

#define D_IN   256
#define D_OUT  96
#define MINN   1e-15f
#define MAXN   0.996f          // (1 - 4e-3)/sqrt(c)
#define ARTC   (1.0f - 1e-7f)  // artanh clip

__device__ __forceinline__ float wsum(float v) {
    v += __shfl_xor(v, 16, 32);
    v += __shfl_xor(v, 8, 32);
    v += __shfl_xor(v, 4, 32);
    v += __shfl_xor(v, 2, 32);
    v += __shfl_xor(v, 1, 32);
    return v;
}

// ---- hyperbolic bias: hb = proj(expmap0(bias)), 96 elements, one block ----
__global__ void bias_kernel(const float* __restrict__ bias, float* __restrict__ hb) {
    __shared__ float red[4];
    int t = threadIdx.x;                       // blockDim.x == 128
    float v = (t < D_OUT) ? bias[t] : 0.0f;
    float ss = wsum(v * v);
    if ((t & 31) == 0) red[t >> 5] = ss;
    __syncthreads();
    float n = fmaxf(sqrtf(red[0] + red[1] + red[2] + red[3]), MINN);
    float p = tanhf(n) / n * v;                // expmap0
    __syncthreads();
    float ps = wsum(p * p);
    if ((t & 31) == 0) red[t >> 5] = ps;
    __syncthreads();
    float pn = fmaxf(sqrtf(red[0] + red[1] + red[2] + red[3]), MINN);
    float pf = (pn > MAXN) ? MAXN / pn : 1.0f; // proj
    if (t < D_OUT) hb[t] = p * pf;
}

// ---- mx = x @ W^T ; one wave per 16-row strip, all 96 output columns ------
__global__ __launch_bounds__(256) void gemm_wmma(const float* __restrict__ x,
                                                 const float* __restrict__ w,
                                                 float* __restrict__ mx, int N) {
    const int nTilesM = (N + 15) >> 4;               // 3125
    const int tileM   = (blockIdx.x << 3) + (threadIdx.x >> 5);
    if (tileM >= nTilesM) return;                    // whole-wave exit: EXEC stays all-1s

    const int lane = threadIdx.x & 31;
    const int half = lane >> 4;                      // selects K pair {0,1} vs {2,3}
    const int mr   = lane & 15;

    int mA = tileM * 16 + mr; if (mA > N - 1) mA = N - 1;   // clamp (N % 16 == 0 here)
    const float* aP = x + (size_t)mA * D_IN + 2 * half;     // A: 16x4 rows of x
    const float* bP = w + (size_t)mr * D_IN + 2 * half;     // B: W^T; + t*16*D_IN + k

    v8f acc[6] = {{}, {}, {}, {}, {}, {}};
#pragma unroll 2
    for (int k = 0; k < D_IN; k += 4) {
        v2f a = *(const v2f*)(aP + k);               // A loaded once per K-step
#pragma unroll
        for (int t = 0; t < 6; ++t) {                // 6 independent accumulators
            v2f b = *(const v2f*)(bP + k + t * 16 * D_IN);
            acc[t] = __builtin_amdgcn_wmma_f32_16x16x4_f32(
                /*neg_a=*/false, a, /*neg_b=*/false, b,
                /*c_mod=*/(short)0, acc[t], /*reuse_a=*/false, /*reuse_b=*/false);
        }
    }

    // D layout: lane holds D[r + 8*half][16*t + mr] in acc[t][r]
    if (tileM * 16 + 15 < N) {                       // full tile: unguarded stores
        float* base = mx + (size_t)(tileM * 16 + 8 * half) * D_OUT + mr;
#pragma unroll
        for (int r = 0; r < 8; ++r) {
            float* o = base + (size_t)r * D_OUT;
#pragma unroll
            for (int t = 0; t < 6; ++t) o[t * 16] = acc[t][r];
        }
    } else {                                         // cold tail path
        for (int r = 0; r < 8; ++r) {
            int m = tileM * 16 + r + 8 * half;
            if (m < N) {
                float* o = mx + (size_t)m * D_OUT + mr;
                for (int t = 0; t < 6; ++t) o[t * 16] = acc[t][r];
            }
        }
    }
}

// ---- HypLinear tail + logmap0 : xt = logmap0(proj(mobius_add(proj(mv), hb)))
__global__ __launch_bounds__(256) void hyp_linear_epilogue(
    const float* __restrict__ x, const float* __restrict__ mx,
    const float* __restrict__ hb, float* __restrict__ xt, int N) {
    int row = (blockIdx.x << 3) + (threadIdx.x >> 5);
    if (row >= N) return;
    int lane = threadIdx.x & 31;

    // ||x_row|| over 256 dims (8/lane)
    const float* xr = x + (size_t)row * D_IN;
    float ss = 0.0f;
#pragma unroll
    for (int j = 0; j < 8; ++j) { float v = xr[lane + 32 * j]; ss = fmaf(v, v, ss); }
    float xn = fmaxf(sqrtf(wsum(ss)), MINN);

    // mobius_matvec tail
    const float* mr = mx + (size_t)row * D_OUT;
    float m0 = mr[lane], m1 = mr[lane + 32], m2 = mr[lane + 64];
    float mn = fmaxf(sqrtf(wsum(m0 * m0 + m1 * m1 + m2 * m2)), MINN);
    float sc = tanhf(mn / xn * atanhf(fminf(xn, ARTC))) / mn;   // mx==0 row -> all zeros
    float h0 = sc * m0, h1 = sc * m1, h2 = sc * m2;
    // proj
    float hn = fmaxf(sqrtf(wsum(h0 * h0 + h1 * h1 + h2 * h2)), MINN);
    float pf = (hn > MAXN) ? MAXN / hn : 1.0f;
    h0 *= pf; h1 *= pf; h2 *= pf;

    // mobius_add(h, hb)
    float b0 = hb[lane], b1 = hb[lane + 32], b2 = hb[lane + 64];
    float x2 = wsum(h0 * h0 + h1 * h1 + h2 * h2);
    float y2 = wsum(b0 * b0 + b1 * b1 + b2 * b2);
    float xy = wsum(h0 * b0 + h1 * b1 + h2 * b2);
    float ca  = 1.0f + 2.0f * xy + y2;
    float cb  = 1.0f - x2;
    float den = fmaxf(1.0f + 2.0f * xy + x2 * y2, MINN);
    float g0 = (ca * h0 + cb * b0) / den;
    float g1 = (ca * h1 + cb * b1) / den;
    float g2 = (ca * h2 + cb * b2) / den;
    // proj
    float gn = fmaxf(sqrtf(wsum(g0 * g0 + g1 * g1 + g2 * g2)), MINN);
    pf = (gn > MAXN) ? MAXN / gn : 1.0f;
    g0 *= pf; g1 *= pf; g2 *= pf;
    // logmap0
    float pn  = fminf(gn, MAXN);
    float lsc = atanhf(fminf(pn, ARTC)) / pn;
    float* o = xt + (size_t)row * D_OUT;
    o[lane] = lsc * g0; o[lane + 32] = lsc * g1; o[lane + 64] = lsc * g2;
}

__global__ void zero_kernel(float* __restrict__ p, size_t n) {
    size_t i = (size_t)blockIdx.x * blockDim.x + threadIdx.x;
    size_t s = (size_t)gridDim.x * blockDim.x;
    for (; i < n; i += s) p[i] = 0.0f;
}

// ---- spmm: support[row] += w_e * xt[col]; one wave per edge, 3 dims/lane ---
__global__ __launch_bounds__(256) void spmm_kernel(
    const int* __restrict__ erow, const int* __restrict__ ecol,
    const float* __restrict__ ew, const float* __restrict__ xt,
    float* __restrict__ sup, int E) {
    int tid  = blockIdx.x * blockDim.x + threadIdx.x;
    int lane = tid & 31;
    int waveStride = (gridDim.x * blockDim.x) >> 5;
    for (int e = tid >> 5; e < E; e += waveStride) {
        int   r = erow[e];
        int   c = ecol[e];
        float w = ew[e];
        const float* src = xt + (size_t)c * D_OUT;
        float*       dst = sup + (size_t)r * D_OUT;
        atomicAdd(dst + lane,      w * src[lane]);
        atomicAdd(dst + lane + 32, w * src[lane + 32]);
        atomicAdd(dst + lane + 64, w * src[lane + 64]);
    }
}

// ---- out = proj(expmap0(relu(logmap0(proj(expmap0(support)))))) ------------
__global__ __launch_bounds__(256) void hyp_agg_act(const float* __restrict__ sup,
                                                   float* __restrict__ out, int N) {
    int row = (blockIdx.x << 3) + (threadIdx.x >> 5);
    if (row >= N) return;
    int lane = threadIdx.x & 31;
    const float* sr = sup + (size_t)row * D_OUT;
    float s0 = sr[lane], s1 = sr[lane + 32], s2 = sr[lane + 64];
    // expmap0
    float sn = fmaxf(sqrtf(wsum(s0 * s0 + s1 * s1 + s2 * s2)), MINN);
    float e  = tanhf(sn) / sn;
    float p0 = e * s0, p1 = e * s1, p2 = e * s2;
    // proj
    float pnr = fmaxf(sqrtf(wsum(p0 * p0 + p1 * p1 + p2 * p2)), MINN);
    float pf  = (pnr > MAXN) ? MAXN / pnr : 1.0f;
    p0 *= pf; p1 *= pf; p2 *= pf;
    // logmap0 + relu
    float pn = fminf(pnr, MAXN);
    float l  = atanhf(fminf(pn, ARTC)) / pn;
    float u0 = fmaxf(l * p0, 0.0f), u1 = fmaxf(l * p1, 0.0f), u2 = fmaxf(l * p2, 0.0f);
    // expmap0
    float un = fmaxf(sqrtf(wsum(u0 * u0 + u1 * u1 + u2 * u2)), MINN);
    float e2 = tanhf(un) / un;
    float h0 = e2 * u0, h1 = e2 * u1, h2 = e2 * u2;
    // proj
    float hn  = fmaxf(sqrtf(wsum(h0 * h0 + h1 * h1 + h2 * h2)), MINN);
    float pf2 = (hn > MAXN) ? MAXN / hn : 1.0f;
    float* o = out + (size_t)row * D_OUT;
    o[lane] = h0 * pf2; o[lane + 32] = h1 * pf2; o[lane + 64] = h2 * pf2;
}

extern "C" void kernel_launch(void* const* d_in, const int* in_sizes, int n_in,
                              void* d_out, int out_size, void* d_ws, size_t ws_size,
                              hipStream_t stream) {
    const float* x      = (const float*)d_in[0];   // [N, 256]
    const float* weight = (const float*)d_in[1];   // [96, 256]
    const float* bias   = (const float*)d_in[2];   // [96]
    const float* ew     = (const float*)d_in[3];   // [E]
    const int*   erow   = (const int*)d_in[4];     // [E]
    const int*   ecol   = (const int*)d_in[5];     // [E]
    const int N = in_sizes[0] / D_IN;              // 50000
    const int E = in_sizes[3];                     // 800000

    float* ws   = (float*)d_ws;
    float* hb   = ws;                               // 128 floats (96 used)
    float* buf0 = ws + 128;                         // mx, later reused as support
    float* buf1 = buf0 + (size_t)N * D_OUT;         // xt
    float* out  = (float*)d_out;

    // 1) hyperbolic bias
    bias_kernel<<<1, 128, 0, stream>>>(bias, hb);

    // 2) fp32 WMMA GEMM: mx = x @ W^T (one wave per 16-row strip)
    int strips     = (N + 15) >> 4;
    int gemmBlocks = (strips + 7) >> 3;             // 8 waves per 256-thread block
    gemm_wmma<<<gemmBlocks, 256, 0, stream>>>(x, weight, buf0, N);

    // 3) HypLinear epilogue -> xt = logmap0(h)
    hyp_linear_epilogue<<<(N + 7) >> 3, 256, 0, stream>>>(x, buf0, hb, buf1, N);

    // 4) zero support, then edge scatter
    zero_kernel<<<2048, 256, 0, stream>>>(buf0, (size_t)N * D_OUT);
    spmm_kernel<<<(E + 7) >> 3, 256, 0, stream>>>(erow, ecol, ew, buf1, buf0, E);

    // 5) HypAgg expmap + HypAct
    hyp_agg_act<<<(N + 7) >> 3, 256, 0, stream>>>(buf0, out, N);
}